// BiTransition_41961830482675
// MI455X (gfx1250) — compile-verified
//
#include <hip/hip_runtime.h>
#include <stdint.h>

// Problem size from the reference.
#define NDIM 8192
#define TPB 256
#define ROW_F4 (NDIM / 4)                 // 2048 float4 per row
#define F4_PER_THREAD (ROW_F4 / TPB)      // 8 float4 per thread

typedef __attribute__((ext_vector_type(4))) unsigned int u32x4;
typedef __attribute__((ext_vector_type(8))) int i32x8;
typedef __attribute__((ext_vector_type(4))) int i32x4;

// TDM builtin exists on both probed toolchains but with different arity.
// The gfx1250 TDM header ships only with the 6-arg (therock) toolchain.
#if defined(__AMDGCN__) && __has_builtin(__builtin_amdgcn_tensor_load_to_lds)
#define USE_TDM 1
#if __has_include(<hip/amd_detail/amd_gfx1250_TDM.h>)
#define TDM_ARGS6 1
#endif
#endif

// One block per row: TDM-stage the 32KB row into LDS, deterministic tree
// row-sum, then scale and store. HBM traffic = read-once + write-once.
__global__ __launch_bounds__(TPB) void rownorm_tdm_kernel(
    const float* __restrict__ src, float* __restrict__ dst) {
  const int row = blockIdx.x;
  const int t = (int)threadIdx.x;

  __shared__ float lds_row[NDIM];      // 32 KB staging for one row
  __shared__ float lds_part[TPB / 32]; // per-wave partial sums
  __shared__ float lds_deg;            // broadcast row sum

  float4 v[F4_PER_THREAD];
  float psum = 0.0f;

#if defined(USE_TDM)
  // --- Tensor Data Mover: DMA one contiguous row (8192 fp32) into LDS. ---
  // Issue from wave 0 only (uniform scalar branch so other waves skip it).
  const int wave = __builtin_amdgcn_readfirstlane(t >> 5);
  if (wave == 0) {
    const uint64_t ga = (uint64_t)(uintptr_t)(src + (size_t)row * NDIM);
    const uint32_t lds_base = (uint32_t)(uintptr_t)&lds_row[0];

    // D# group 0 (128b): count=1 valid user descriptor; lds_addr; global
    // byte address [56:0]; type=2 ("image") in bits [127:126].
    u32x4 g0 = {
        1u,                                               // count=1
        lds_base,                                         // lds_addr (bytes)
        (uint32_t)ga,                                     // global_addr[31:0]
        ((uint32_t)(ga >> 32) & 0x01FFFFFFu) | 0x80000000u // addr[56:32]|type=2
    };
    // D# group 1 (256b): no multicast, data_size=2 (4B), no pad/iterate.
    // 1-D tile: tile_dim0 = 8192 elements, tensor_dim0 = 8192,
    // tensor_dim0_stride = 8192 (row-major, unused for a single line).
    i32x8 g1 = {
        (int)0x00020000u,                  // [17:16] data_size = 4 bytes
        (int)((NDIM & 0xFFFF) << 16),      // [63:48] tensor_dim0[15:0]
        (int)0x00010000u,                  // [79:64]=dim0 hi=0, [95:80] tensor_dim1=1
        (int)((NDIM & 0xFFFF) << 16),      // [111:96]=dim1 hi=0, [127:112] tile_dim0
        1,                                 // [143:128] tile_dim1=1, tile_dim2=0
        NDIM,                              // [191:160] tensor_dim0_stride lo
        0,                                 // stride hi / tensor_dim1_stride lo
        0
    };
    i32x4 z4 = {0, 0, 0, 0};
#if defined(TDM_ARGS6)
    i32x8 z8 = {0, 0, 0, 0, 0, 0, 0, 0};
    __builtin_amdgcn_tensor_load_to_lds(g0, g1, z4, z4, z8, 0);
#else
    __builtin_amdgcn_tensor_load_to_lds(g0, g1, z4, z4, 0);
#endif
    __builtin_amdgcn_s_wait_tensorcnt(0); // DMA complete for this wave
  }
  __syncthreads(); // publish LDS row to all 8 waves

  const float4* lrow = (const float4*)lds_row;
#pragma unroll
  for (int k = 0; k < F4_PER_THREAD; ++k) {
    v[k] = lrow[t + k * TPB]; // 16B lane stride -> conflict-free ds_load_b128
    psum += (v[k].x + v[k].y) + (v[k].z + v[k].w);
  }
#else
  // Fallback (no TDM builtin): direct coalesced b128 global loads.
  const float4* grow = (const float4*)(src + (size_t)row * NDIM);
#pragma unroll
  for (int k = 0; k < F4_PER_THREAD; ++k) {
    v[k] = grow[t + k * TPB];
    psum += (v[k].x + v[k].y) + (v[k].z + v[k].w);
  }
#endif

  // Deterministic block reduction: wave32 shuffle tree, then serial 8-way.
#pragma unroll
  for (int off = 16; off > 0; off >>= 1)
    psum += __shfl_down(psum, (unsigned)off, 32);
  if ((t & 31) == 0) lds_part[t >> 5] = psum;
  __syncthreads();
  if (t == 0) {
    float s = 0.0f;
#pragma unroll
    for (int w = 0; w < TPB / 32; ++w) s += lds_part[w];
    lds_deg = s;
  }
  __syncthreads();
  const float deg = lds_deg;

  // Scale and stream out (coalesced global_store_b128).
  float4* orow = (float4*)(dst + (size_t)row * NDIM);
#pragma unroll
  for (int k = 0; k < F4_PER_THREAD; ++k) {
    float4 o;
    o.x = v[k].x / deg;
    o.y = v[k].y / deg;
    o.z = v[k].z / deg;
    o.w = v[k].w / deg;
    orow[t + k * TPB] = o;
  }
}

extern "C" void kernel_launch(void* const* d_in, const int* in_sizes, int n_in,
                              void* d_out, int out_size, void* d_ws,
                              size_t ws_size, hipStream_t stream) {
  (void)in_sizes; (void)n_in; (void)out_size; (void)d_ws; (void)ws_size;

  const float* graph0 = (const float*)d_in[0];
  const float* graph1 = (const float*)d_in[1];
  float* out = (float*)d_out;
  const size_t elems = (size_t)NDIM * NDIM;

  // Output 0: identity pass-through of graph0 (pure D2D copy, capture-safe).
  hipMemcpyAsync(out, graph0, elems * sizeof(float),
                 hipMemcpyDeviceToDevice, stream);

  // Output 1: row-normalized graph1.
  rownorm_tdm_kernel<<<NDIM, TPB, 0, stream>>>(graph1, out + elems);
}